// VQCLayer_62388694941856
// MI455X (gfx1250) — compile-verified
//
#include <hip/hip_runtime.h>

typedef float v2f __attribute__((ext_vector_type(2)));
typedef float v8f __attribute__((ext_vector_type(8)));

#define NQ 16
#define NL 3
#define PI_F 3.14159265f
#define LB 12            // local qubits per pass
#define CHUNK 4096       // 2^LB amplitudes per LDS chunk
#define NCHUNK 16        // 65536 / 4096
#define THREADS 256      // 8 wave32s

// Entry of the 16x16 tensor-product of four real Ry(theta) matrices.
// row bit a, col bit b per qubit: (a==b)->cos, (1,0)->sin, (0,1)->-sin
__device__ __forceinline__ float uentry(int r, int k, int q0,
                                        const float* gc, const float* gs) {
  float v = 1.0f;
#pragma unroll
  for (int q = 0; q < 4; ++q) {
    int a = (r >> q) & 1, b = (k >> q) & 1;
    float c = gc[q0 + q], s = gs[q0 + q];
    v *= (a == b) ? c : (a ? s : -s);
  }
  return v;
}

__global__ __launch_bounds__(THREADS)
void vqc_kernel(const float* __restrict__ x, const float* __restrict__ params,
                float* __restrict__ out, float2* __restrict__ ws) {
  __shared__ float sRe[CHUNK];
  __shared__ float sIm[CHUNK];
  __shared__ float gc[NQ], gs[NQ], hph[NQ];
  __shared__ float red[THREADS];

  const int tid  = threadIdx.x;
  const int lane = tid & 31;
  const int wave = tid >> 5;
  const int hi   = lane >> 4;   // half-wave select (K/M split in WMMA layouts)
  const int nn   = lane & 15;   // row (A) / column (B,D) index

  float2* st = ws + (size_t)blockIdx.x * 65536u;  // 512KB state slice per block

  for (int b = blockIdx.x; b < 512; b += gridDim.x) {
    const float* data = x + b * NQ;
    float acc0 = 0.f, acc1 = 0.f, acc2 = 0.f;

    for (int layer = 0; layer < NL; ++layer) {
      const int base = layer * 2 * NQ;
      __syncthreads();
      if (tid < NQ) {
        float th = 0.5f * (data[tid] * PI_F + params[base + tid]);
        gc[tid]  = __cosf(th);
        gs[tid]  = __sinf(th);
        hph[tid] = 0.5f * params[base + NQ + tid];  // Rz half-angle
      }
      __syncthreads();

      // ---------- Pass A: local qubits 0..11 (contiguous chunks) ----------
      for (int chunk = 0; chunk < NCHUNK; ++chunk) {
        const int gbase = chunk << LB;
        if (layer == 0) {  // synthesize |0...0>, no global read
          for (int e = tid; e < CHUNK; e += THREADS) {
            sRe[e] = (chunk == 0 && e == 0) ? 1.0f : 0.0f;
            sIm[e] = 0.0f;
          }
        } else {
          for (int e = tid; e < CHUNK; e += THREADS) {
            float2 a = st[gbase + e];
            sRe[e] = a.x; sIm[e] = a.y;
          }
        }
        __syncthreads();

        // Ry groups {0-3},{4-7},{8-11}: S <- U * S  via f32 WMMA (re & im)
        for (int g = 0; g < 3; ++g) {
          const int q0 = 4 * g;
          v2f A[4];  // U[:,4t:4t+4] in A-layout (16x4 f32 = 2 VGPRs)
#pragma unroll
          for (int t = 0; t < 4; ++t) {
            int k0 = 4 * t + 2 * hi;
            A[t].x = uentry(nn, k0,     q0, gc, gs);
            A[t].y = uentry(nn, k0 + 1, q0, gc, gs);
          }
          const int lowmask = (1 << q0) - 1;
          for (int tile = wave; tile < 16; tile += 8) {  // uniform per wave
            int col = tile * 16 + nn;
            int jc  = (col & lowmask) | ((col >> q0) << (q0 + 4));
            v8f cR = {}; v8f cI = {};
#pragma unroll
            for (int t = 0; t < 4; ++t) {
              int k0 = 4 * t + 2 * hi;
              int j0 = jc | (k0 << q0);
              int j1 = jc | ((k0 + 1) << q0);
              v2f bR = { sRe[j0], sRe[j1] };
              v2f bI = { sIm[j0], sIm[j1] };
              cR = __builtin_amdgcn_wmma_f32_16x16x4_f32(
                     false, A[t], false, bR, (short)0, cR, false, false);
              cI = __builtin_amdgcn_wmma_f32_16x16x4_f32(
                     false, A[t], false, bI, (short)0, cI, false, false);
            }
#pragma unroll
            for (int r = 0; r < 8; ++r) {
              int j = jc | ((r + 8 * hi) << q0);
              sRe[j] = cR[r];
              sIm[j] = cI[r];
            }
          }
          __syncthreads();
        }

        // write-back folding CNOT(0,1)..(10,11) == prefix-XOR over 12 bits
        for (int e = tid; e < CHUNK; e += THREADS) {
          int y = e;
          y ^= y << 1; y ^= y << 2; y ^= y << 4; y ^= y << 8;
          y &= CHUNK - 1;
          float2 o; o.x = sRe[e]; o.y = sIm[e];
          st[gbase + y] = o;
        }
        __syncthreads();
      }
      __threadfence();
      __syncthreads();

      // ---------- Pass B: local qubits {0..6, 11..15} (strided gather) ----------
      const bool lastL = (layer == NL - 1);
      for (int chunk = 0; chunk < NCHUNK; ++chunk) {
        // local j: bits0-6 <-> q0-6, bits7-11 <-> q11-15; chunk <-> q7-10
        for (int e = tid; e < CHUNK; e += THREADS) {
          int gi = (e & 127) | (chunk << 7) | ((e >> 7) << 11);
          float2 a = st[gi];
          sRe[e] = a.x; sIm[e] = a.y;
        }
        __syncthreads();

        // Ry(q12..q15) on local bits 8..11 via WMMA
        {
          v2f A[4];
#pragma unroll
          for (int t = 0; t < 4; ++t) {
            int k0 = 4 * t + 2 * hi;
            A[t].x = uentry(nn, k0,     12, gc, gs);
            A[t].y = uentry(nn, k0 + 1, 12, gc, gs);
          }
          for (int tile = wave; tile < 16; tile += 8) {
            int col = tile * 16 + nn;  // occupies local bits 0..7
            v8f cR = {}; v8f cI = {};
#pragma unroll
            for (int t = 0; t < 4; ++t) {
              int k0 = 4 * t + 2 * hi;
              v2f bR = { sRe[col | (k0 << 8)], sRe[col | ((k0 + 1) << 8)] };
              v2f bI = { sIm[col | (k0 << 8)], sIm[col | ((k0 + 1) << 8)] };
              cR = __builtin_amdgcn_wmma_f32_16x16x4_f32(
                     false, A[t], false, bR, (short)0, cR, false, false);
              cI = __builtin_amdgcn_wmma_f32_16x16x4_f32(
                     false, A[t], false, bI, (short)0, cI, false, false);
            }
#pragma unroll
            for (int r = 0; r < 8; ++r) {
              int j = col | ((r + 8 * hi) << 8);
              sRe[j] = cR[r];
              sIm[j] = cI[r];
            }
          }
        }
        __syncthreads();

        // CNOT(11,12)(12,13)(13,14)(14,15)(15,0) as index perm + Rz phase
        // at destination; last layer: accumulate <Z0..2> instead of store.
        for (int e = tid; e < CHUNK; e += THREADS) {
          int n8  = ((e >> 8) ^ (e >> 7)) & 1;       // q12 ^= q11
          int n9  = ((e >> 9)  & 1) ^ n8;            // q13 ^= q12'
          int n10 = ((e >> 10) & 1) ^ n9;            // q14 ^= q13'
          int n11 = ((e >> 11) & 1) ^ n10;           // q15 ^= q14'
          int n0  = (e & 1) ^ n11;                   // q0  ^= q15'
          int d  = (e & 0x0FE) | n0 | (n8 << 8) | (n9 << 9) |
                   (n10 << 10) | (n11 << 11);
          int gi = (d & 127) | (chunk << 7) | ((d >> 7) << 11);
          float re = sRe[e], im = sIm[e];
          if (!lastL) {
            float ang = 0.f;
#pragma unroll
            for (int q = 0; q < NQ; ++q)
              ang += ((gi >> q) & 1) ? hph[q] : -hph[q];
            float ca = __cosf(ang), sa = __sinf(ang);
            float2 o; o.x = re * ca - im * sa; o.y = re * sa + im * ca;
            st[gi] = o;
          } else {
            float p = re * re + im * im;   // Rz phase drops out of |amp|^2
            acc0 += (gi & 1) ? -p : p;
            acc1 += (gi & 2) ? -p : p;
            acc2 += (gi & 4) ? -p : p;
          }
        }
        __syncthreads();
      }
      if (!lastL) { __threadfence(); __syncthreads(); }
    }

    // block-wide reduction of the three <Z_q> partials
#pragma unroll
    for (int c = 0; c < 3; ++c) {
      float v = (c == 0) ? acc0 : (c == 1) ? acc1 : acc2;
      __syncthreads();
      red[tid] = v;
      __syncthreads();
      for (int s = THREADS / 2; s > 0; s >>= 1) {
        if (tid < s) red[tid] += red[tid + s];
        __syncthreads();
      }
      if (tid == 0) out[b * 3 + c] = red[0];
    }
    __syncthreads();
  }
}

extern "C" void kernel_launch(void* const* d_in, const int* in_sizes, int n_in,
                              void* d_out, int out_size, void* d_ws, size_t ws_size,
                              hipStream_t stream) {
  (void)in_sizes; (void)n_in; (void)out_size;
  const float* x = (const float*)d_in[0];       // (512, 16) f32
  const float* p = (const float*)d_in[1];       // (96,) f32
  float* out = (float*)d_out;                   // (512, 3) f32

  // 512KB of state per concurrently-active block; clamp grid to fit ws.
  const size_t per_block = (size_t)65536 * sizeof(float2);
  int nb = 128;                                  // power-of-two divisor of 512
  while (nb > 1 && (size_t)nb * per_block > ws_size) nb >>= 1;

  vqc_kernel<<<dim3(nb), dim3(THREADS), 0, stream>>>(x, p, out, (float2*)d_ws);
}